// GIN_81570018885850
// MI455X (gfx1250) — compile-verified
//
#include <hip/hip_runtime.h>
#include <cstdint>
#include <cstddef>

// ---------------------------------------------------------------------------
// GIN forward for MI455X (gfx1250, wave32).
//  - scatter kernel: 1 wave per edge, float4 gather + global_atomic_add_f32
//    (X and S are L2-resident: 25.6MB each vs 192MB L2).
//  - fused MLP kernel: Z=(1+eps)X+S -> GEMM1(+b1,ReLU) -> GEMM2(+b2) +
//    residual, using V_WMMA_F32_16X16X4_F32 (fp32-exact, matches reference).
//    W staged in LDS via GLOBAL_LOAD_ASYNC_TO_LDS_B128 when available
//    (ASYNCcnt path), Z/H tile in LDS; 103KB dynamic LDS per block,
//    enabled by CDNA5's 320KB/WGP LDS.
// ---------------------------------------------------------------------------

typedef float v2f __attribute__((ext_vector_type(2)));
typedef float v8f __attribute__((ext_vector_type(8)));
typedef int   v4i __attribute__((vector_size(4 * sizeof(int))));

typedef __attribute__((address_space(1))) v4i gv4i;   // global int32x4
typedef __attribute__((address_space(3))) v4i lv4i;   // LDS int32x4

#if defined(__has_builtin)
#  if __has_builtin(__builtin_amdgcn_global_load_async_to_lds_b128)
#    define HAVE_ASYNC_LDS 1
#  endif
#endif
#ifndef HAVE_ASYNC_LDS
#  define HAVE_ASYNC_LDS 0
#endif

#define ND       128
#define ZSTRIDE  132   // 4-bank rotation/row -> 16 A-frag rows hit 16 distinct bank groups
#define WSTRIDE  136   // 2*136 mod 64 = 16 -> half-wave row pairs of B-frag disjoint banks
#define ROWS_PB   64   // node rows per block (4 waves x 16 rows)
#define THREADS  128

// ---------------------------- scatter-add ----------------------------------
__global__ void __launch_bounds__(256)
gin_scatter(const float* __restrict__ X,
            const long long* __restrict__ src,
            const long long* __restrict__ dst,
            float* __restrict__ S, int E) {
  int tid  = blockIdx.x * blockDim.x + threadIdx.x;
  int edge = tid >> 5;        // one wave32 per edge: 32 lanes x float4 = 128 floats
  int lane = tid & 31;
  if (edge >= E) return;
  int s = (int)src[edge];
  int d = (int)dst[edge];
  const float4 v = *reinterpret_cast<const float4*>(X + (size_t)s * ND + lane * 4);
  float* o = S + (size_t)d * ND + lane * 4;
  __hip_atomic_fetch_add(o + 0, v.x, __ATOMIC_RELAXED, __HIP_MEMORY_SCOPE_AGENT);
  __hip_atomic_fetch_add(o + 1, v.y, __ATOMIC_RELAXED, __HIP_MEMORY_SCOPE_AGENT);
  __hip_atomic_fetch_add(o + 2, v.z, __ATOMIC_RELAXED, __HIP_MEMORY_SCOPE_AGENT);
  __hip_atomic_fetch_add(o + 3, v.w, __ATOMIC_RELAXED, __HIP_MEMORY_SCOPE_AGENT);
}

// ---- weight staging: global -> LDS, async path when toolchain exposes it ---
__device__ __forceinline__ void stage_weights(const float* __restrict__ Wg,
                                              float* __restrict__ Wb, int t) {
#if HAVE_ASYNC_LDS
  for (int c = 0; c < ND; c += 4) {
    // 16B chunks, both addresses 16B aligned; no VGPR round-trip (ASYNCcnt).
    __builtin_amdgcn_global_load_async_to_lds_b128(
        (gv4i*)(Wg + (size_t)t * ND + c),
        (lv4i*)(Wb + t * WSTRIDE + c), /*offset=*/0, /*cpol=*/0);
  }
  asm volatile("s_wait_asynccnt 0" ::: "memory");
#else
  for (int c = 0; c < ND; c += 4) {
    float4 w = *reinterpret_cast<const float4*>(Wg + (size_t)t * ND + c);
    *reinterpret_cast<float4*>(Wb + t * WSTRIDE + c) = w;
  }
#endif
}

// ------------------------- fused MLP (WMMA f32) ----------------------------
__global__ void __launch_bounds__(THREADS)
gin_fused_mlp(const float* __restrict__ Xin, const float* __restrict__ S,
              const float* __restrict__ W1, const float* __restrict__ b1,
              const float* __restrict__ W2, const float* __restrict__ b2,
              const float* __restrict__ eps, int layer,
              float* __restrict__ Xout, int N) {
  extern __shared__ float smem[];
  float* Wb = smem;                    // [128][WSTRIDE] staged weight matrix
  float* Zt = smem + ND * WSTRIDE;     // [64][ZSTRIDE] Z tile, reused as H tile

  const int t     = threadIdx.x;
  const int wave  = t >> 5;
  const int lane  = t & 31;
  const int row16 = lane & 15;         // M / N index within 16x16 tile
  const int khalf = lane >> 4;         // which K-pair half this lane holds
  const int base  = blockIdx.x * ROWS_PB;
  const float e1  = 1.0f + eps[layer];

  // hint: W2 will be needed after GEMM1 (global_prefetch_b8)
  __builtin_prefetch(W2 + (size_t)t * ND, 0, 0);

  // ---- stage W1 into LDS: Wb[k][n], row-major with padded stride ----
  stage_weights(W1, Wb, t);

  // ---- compute Z tile: Z = (1+eps)*X + S (zero-fill OOB rows) ----
  {
    int r  = t >> 1;
    int c0 = (t & 1) * 64;
    int rg = base + r;
    for (int c = 0; c < 64; c += 4) {
      float4 z;
      if (rg < N) {
        float4 x = *reinterpret_cast<const float4*>(Xin + (size_t)rg * ND + c0 + c);
        float4 s = *reinterpret_cast<const float4*>(S   + (size_t)rg * ND + c0 + c);
        z.x = __builtin_fmaf(e1, x.x, s.x);
        z.y = __builtin_fmaf(e1, x.y, s.y);
        z.z = __builtin_fmaf(e1, x.z, s.z);
        z.w = __builtin_fmaf(e1, x.w, s.w);
      } else {
        z = make_float4(0.f, 0.f, 0.f, 0.f);
      }
      *reinterpret_cast<float4*>(Zt + r * ZSTRIDE + c0 + c) = z;
    }
  }
  __syncthreads();

  // Each wave owns a 16-row slab: rows [wave*16, wave*16+16).
  const int arow = wave * 16 + row16;

  // ---- preload A fragments for GEMM1 (Z slab fully into registers) ----
  // A 16x4 f32 layout: lane row16 holds K = {2*khalf, 2*khalf+1} per step.
  v2f afrag[32];
#pragma unroll
  for (int k4 = 0; k4 < 32; ++k4)
    afrag[k4] = *reinterpret_cast<const v2f*>(Zt + arow * ZSTRIDE + 4 * k4 + 2 * khalf);

  // ---- GEMM1: H = relu(Z @ W1 + b1); write H into own slab of Zt ----
  for (int nt = 0; nt < 8; ++nt) {
    const int col = nt * 16 + row16;
    const float bias = b1[col];
    v8f acc;
#pragma unroll
    for (int v = 0; v < 8; ++v) acc[v] = bias;   // bias depends only on N=col
#pragma unroll
    for (int k4 = 0; k4 < 32; ++k4) {
      const int kr = 4 * k4 + 2 * khalf;
      v2f b;
      b.x = Wb[kr * WSTRIDE + col];
      b.y = Wb[(kr + 1) * WSTRIDE + col];
      acc = __builtin_amdgcn_wmma_f32_16x16x4_f32(
          /*neg_a=*/false, afrag[k4], /*neg_b=*/false, b,
          /*c_mod=*/(short)0, acc, /*reuse_a=*/false, /*reuse_b=*/false);
    }
#pragma unroll
    for (int v = 0; v < 8; ++v) {
      // C/D element v -> (M = v + 8*khalf, N = col); slab is wave-private.
      Zt[(wave * 16 + v + 8 * khalf) * ZSTRIDE + col] = fmaxf(acc[v], 0.0f);
    }
  }
  __syncthreads();          // all waves done reading W1 from Wb

  // ---- stage W2 into LDS (overwrite Wb) ----
  stage_weights(W2, Wb, t);
  __syncthreads();

  // ---- preload A fragments for GEMM2 (H slab, wave-private rows) ----
#pragma unroll
  for (int k4 = 0; k4 < 32; ++k4)
    afrag[k4] = *reinterpret_cast<const v2f*>(Zt + arow * ZSTRIDE + 4 * k4 + 2 * khalf);

  // ---- GEMM2: Xout = H @ W2 + b2 + Xin (residual), guarded store ----
  for (int nt = 0; nt < 8; ++nt) {
    const int col = nt * 16 + row16;
    const float bias = b2[col];
    v8f acc;
#pragma unroll
    for (int v = 0; v < 8; ++v) acc[v] = bias;
#pragma unroll
    for (int k4 = 0; k4 < 32; ++k4) {
      const int kr = 4 * k4 + 2 * khalf;
      v2f b;
      b.x = Wb[kr * WSTRIDE + col];
      b.y = Wb[(kr + 1) * WSTRIDE + col];
      acc = __builtin_amdgcn_wmma_f32_16x16x4_f32(
          false, afrag[k4], false, b, (short)0, acc, false, false);
    }
#pragma unroll
    for (int v = 0; v < 8; ++v) {
      int rg = base + wave * 16 + v + 8 * khalf;
      if (rg < N)
        Xout[(size_t)rg * ND + col] = acc[v] + Xin[(size_t)rg * ND + col];
    }
  }
}

// ------------------------------- launcher ----------------------------------
extern "C" void kernel_launch(void* const* d_in, const int* in_sizes, int n_in,
                              void* d_out, int out_size, void* d_ws, size_t ws_size,
                              hipStream_t stream) {
  const float*     X   = (const float*)d_in[0];
  const long long* ei  = (const long long*)d_in[1];   // int64 edge_index [2,E]
  const float*     eps = (const float*)d_in[2];
  const float*     W1  = (const float*)d_in[3];
  const float*     b1  = (const float*)d_in[4];
  const float*     W2  = (const float*)d_in[5];
  const float*     b2  = (const float*)d_in[6];
  float*           out = (float*)d_out;

  const int N = in_sizes[0] / ND;        // 50000
  const int E = in_sizes[1] / 2;         // 800000
  const int L = in_sizes[2];             // 3

  float* S    = (float*)d_ws;            // [N, ND] neighbor sums
  float* buf0 = S + (size_t)N * ND;      // [N, ND] ping-pong buffer

  const size_t shmem = (size_t)(ND * WSTRIDE + ROWS_PB * ZSTRIDE) * sizeof(float);
  const int fused_blocks   = (N + ROWS_PB - 1) / ROWS_PB;
  const int scatter_blocks = (int)(((long long)E * 32 + 255) / 256);

  const float* cur = X;
  for (int l = 0; l < L; ++l) {
    // next-buffer schedule: l0 -> d_out, l1 -> buf0, l2 -> d_out
    float* nxt = (l == 1) ? buf0 : out;

    (void)hipMemsetAsync(S, 0, (size_t)N * ND * sizeof(float), stream);
    gin_scatter<<<scatter_blocks, 256, 0, stream>>>(cur, ei, ei + E, S, E);
    gin_fused_mlp<<<fused_blocks, THREADS, shmem, stream>>>(
        cur, S, W1 + (size_t)l * ND * ND, b1 + (size_t)l * ND,
        W2 + (size_t)l * ND * ND, b2 + (size_t)l * ND, eps, l, nxt, N);
    cur = nxt;
  }
}